// GRU_RGCN_76003741270422
// MI455X (gfx1250) — compile-verified
//
#include <hip/hip_runtime.h>
#include <math.h>

typedef __attribute__((ext_vector_type(2))) float v2f;
typedef __attribute__((ext_vector_type(8))) float v8f;

constexpr int kB  = 16;
constexpr int kN  = 256;
constexpr int kD  = 512;
constexpr int kR  = 5;
constexpr int kE  = 1024;
constexpr int kVG = 24000;
constexpr int kVS = 13000;

constexpr int kG  = kR + 1;                  // 6 GEMMs per step (5 relations + W0)
constexpr int kTM = kN / 16;                 // 16 tiles along M
constexpr int kTN = kD / 16;                 // 32 tiles along N
constexpr int kTilesPerGemm = kTM * kTN;     // 512
constexpr int kGemmWaves    = kG * kTilesPerGemm;   // 3072
constexpr int kWavesPerBlk  = 8;                    // 256 threads (8 wave32)
constexpr int kGemmBlocks   = kGemmWaves / kWavesPerBlk; // 384

constexpr int kChunkRows = 512;              // gate GEMV rows per chunk
constexpr int kChunks    = (kN * kD) / kChunkRows;  // 256

// ---------------------------------------------------------------------------
// H[g] = xs @ Wg for g in 0..5  (Wg = conv_W[g] for g<5, else W0)
// One wave per 16x16 output tile, K-loop of 128 x V_WMMA_F32_16X16X4_F32.
// Lane layouts per CDNA5 ISA 7.12.2:
//   A 16x4 : lane l -> M = l%16 ; vgpr v holds K = v + 2*(l/16)
//   B 4x16 : lane l -> N = l%16 ; vgpr v holds K = v + 2*(l/16)
//   C 16x16: lane l -> N = l%16 ; vgpr v holds M = v + 8*(l/16)
// ---------------------------------------------------------------------------
__global__ void gemm6_wmma_kernel(const float* __restrict__ xs,     // [N,D]
                                  const float* __restrict__ convW,  // [R,D,D]
                                  const float* __restrict__ W0,     // [D,D]
                                  float* __restrict__ H)            // [6,N,D]
{
    const int wave = blockIdx.x * kWavesPerBlk + (threadIdx.x >> 5);
    const int lane = threadIdx.x & 31;
    const int g    = wave / kTilesPerGemm;          // uniform per wave
    const int t    = wave % kTilesPerGemm;
    const int m0   = (t / kTN) * 16;
    const int n0   = (t % kTN) * 16;
    const int row  = lane & 15;
    const int half = lane >> 4;

    const float* __restrict__ Wm =
        (g < kR) ? (convW + (size_t)g * kD * kD) : W0;
    const float* __restrict__ Arow = xs + (size_t)(m0 + row) * kD + 2 * half;

    v8f acc = {};
#pragma unroll 4
    for (int k = 0; k < kD; k += 4) {
        v2f a, b;
        a.x = Arow[k];
        a.y = Arow[k + 1];
        b.x = Wm[(size_t)(k + 2 * half) * kD + n0 + row];
        b.y = Wm[(size_t)(k + 2 * half + 1) * kD + n0 + row];
#if __has_builtin(__builtin_amdgcn_wmma_f32_16x16x4_f32)
        acc = __builtin_amdgcn_wmma_f32_16x16x4_f32(
            /*neg_a=*/false, a, /*neg_b=*/false, b,
            /*c_mod=*/(short)0, acc, /*reuse_a=*/false, /*reuse_b=*/false);
#else
        // Scalar fallback computing the identical tile layout.
#pragma unroll
        for (int v = 0; v < 8; ++v) {
            const int M = m0 + v + 8 * half;
            const int Nn = n0 + row;
            float s = acc[v];
            s += xs[(size_t)M * kD + k + 0] * Wm[(size_t)(k + 0) * kD + Nn];
            s += xs[(size_t)M * kD + k + 1] * Wm[(size_t)(k + 1) * kD + Nn];
            s += xs[(size_t)M * kD + k + 2] * Wm[(size_t)(k + 2) * kD + Nn];
            s += xs[(size_t)M * kD + k + 3] * Wm[(size_t)(k + 3) * kD + Nn];
            acc[v] = s;
        }
#endif
    }

    float* __restrict__ O = H + (size_t)g * kN * kD;
#pragma unroll
    for (int v = 0; v < 8; ++v)
        O[(size_t)(m0 + v + 8 * half) * kD + n0 + row] = acc[v];
}

// ---------------------------------------------------------------------------
// dis[r][n] = rsqrt(1 + #edges with dst==n).  Deterministic via LDS int atomics.
// ---------------------------------------------------------------------------
__global__ void degree_kernel(const int* __restrict__ ei,   // [R,2,E] step slice
                              float* __restrict__ dis)      // [R,N]
{
    __shared__ int cnt[kN];
    const int r = blockIdx.x;
    const int n = threadIdx.x;          // 256 threads
    cnt[n] = 0;
    __syncthreads();
    const int* __restrict__ dst = ei + ((size_t)r * 2 + 1) * kE;
    for (int e = n; e < kE; e += kN)
        atomicAdd(&cnt[dst[e]], 1);
    __syncthreads();
    dis[r * kN + n] = rsqrtf((float)(cnt[n] + 1));
}

// ---------------------------------------------------------------------------
// proposed[n][d] = sum_r [ sum_{e:dst==n} H_r[src_e][d]*dis_r[src]*dis_r[n]
//                          + H_r[n][d]*dis_r[n]^2 ] + H0[n][d]
// One block per node n; deterministic fixed-order edge scan (uniform branch).
// ---------------------------------------------------------------------------
__global__ void aggregate_kernel(const int* __restrict__ ei,     // [R,2,E]
                                 const float* __restrict__ dis,  // [R,N]
                                 const float* __restrict__ H,    // [6,N,D]
                                 float* __restrict__ proposed)   // [N,D]
{
    const int n = blockIdx.x;
    const int d = threadIdx.x;          // 512 threads
    float acc = H[(size_t)kR * kN * kD + (size_t)n * kD + d];   // x @ W0 term
#pragma unroll 1
    for (int r = 0; r < kR; ++r) {
        const float* __restrict__ Hr   = H + (size_t)r * kN * kD;
        const int*   __restrict__ src  = ei + (size_t)r * 2 * kE;
        const int*   __restrict__ dst  = src + kE;
        const float* __restrict__ disr = dis + r * kN;
        const float dn = disr[n];
        float a = 0.f;
        for (int e = 0; e < kE; ++e) {
            if (dst[e] == n) {                        // uniform across block
                const int s = src[e];
                a += Hr[(size_t)s * kD + d] * (disr[s] * dn);
            }
        }
        acc += a + Hr[(size_t)n * kD + d] * (dn * dn);
    }
    proposed[(size_t)n * kD + d] = acc;
}

// ---------------------------------------------------------------------------
// Gate GEMV stage 1: partial[c][j] = sum_{i in chunk c} xs_flat[i]*gate_W[i][j]
// 256 chunks x 512 rows; streams the 256MB gate_W fully coalesced.
// ---------------------------------------------------------------------------
__global__ void gate_partial_kernel(const float* __restrict__ xs,     // [N*D]
                                    const float* __restrict__ gateW,  // [N*D,D]
                                    float* __restrict__ partial)      // [256,D]
{
    __shared__ float xloc[kChunkRows];
    const int c = blockIdx.x;
    const int j = threadIdx.x;          // 512 threads
    const int i0 = c * kChunkRows;
    xloc[j] = xs[i0 + j];
    __syncthreads();
    float acc = 0.f;
#pragma unroll 4
    for (int i = 0; i < kChunkRows; ++i)
        acc += xloc[i] * gateW[(size_t)(i0 + i) * kD + j];
    partial[(size_t)c * kD + j] = acc;
}

// ---------------------------------------------------------------------------
// Gate stage 2: gate[j] = sigmoid(sum_c partial[c][j] + sum_k mem0[k]*gate_U[k][j])
// ---------------------------------------------------------------------------
__global__ void gate_finalize_kernel(const float* __restrict__ partial,
                                     const float* __restrict__ mem,    // row 0
                                     const float* __restrict__ gateU,  // [D,D]
                                     float* __restrict__ gate)         // [D]
{
    const int j = threadIdx.x;          // 512 threads, one block
    float acc = 0.f;
#pragma unroll 4
    for (int c = 0; c < kChunks; ++c)
        acc += partial[(size_t)c * kD + j];
    for (int k = 0; k < kD; ++k)
        acc += mem[k] * gateU[(size_t)k * kD + j];
    gate[j] = 1.f / (1.f + expf(-acc));
}

// ---------------------------------------------------------------------------
// mem = gate*proposed + (1-gate)*mem   (elementwise, gate broadcast over rows)
// ---------------------------------------------------------------------------
__global__ void mem_update_kernel(const float* __restrict__ gate,
                                  const float* __restrict__ proposed,
                                  float* __restrict__ mem)
{
    const int idx = blockIdx.x * blockDim.x + threadIdx.x;   // N*D
    const int d = idx & (kD - 1);
    const float g = gate[d];
    mem[idx] = g * proposed[idx] + (1.f - g) * mem[idx];
}

// ---------------------------------------------------------------------------
// logits[j] = b[j] + sum_k relu(mem0[k]) * W[k][j]     (W is [D,V] row-major)
// ---------------------------------------------------------------------------
__global__ void logits_kernel(const float* __restrict__ mem,  // row 0 of mem
                              const float* __restrict__ W,
                              const float* __restrict__ b,
                              float* __restrict__ logits, int V)
{
    __shared__ float h0[kD];
    const int tid = threadIdx.x;        // 256 threads
    for (int k = tid; k < kD; k += 256)
        h0[k] = fmaxf(mem[k], 0.f);
    __syncthreads();
    const int j = blockIdx.x * 256 + tid;
    if (j >= V) return;
    float acc = b[j];
#pragma unroll 4
    for (int k = 0; k < kD; ++k)
        acc += h0[k] * W[(size_t)k * V + j];
    logits[j] = acc;
}

// ---------------------------------------------------------------------------
// out[j] = logits[j] - max - log(sum exp(logits - max));  one block, 1024 thr.
// ---------------------------------------------------------------------------
__global__ void logsoftmax_kernel(const float* __restrict__ logits,
                                  float* __restrict__ out, int V)
{
    __shared__ float red[1024];
    const int tid = threadIdx.x;
    float m = -__builtin_inff();
    for (int j = tid; j < V; j += 1024) m = fmaxf(m, logits[j]);
    red[tid] = m;
    __syncthreads();
    for (int s = 512; s > 0; s >>= 1) {
        if (tid < s) red[tid] = fmaxf(red[tid], red[tid + s]);
        __syncthreads();
    }
    const float mx = red[0];
    __syncthreads();
    float sum = 0.f;
    for (int j = tid; j < V; j += 1024) sum += expf(logits[j] - mx);
    red[tid] = sum;
    __syncthreads();
    for (int s = 512; s > 0; s >>= 1) {
        if (tid < s) red[tid] += red[tid + s];
        __syncthreads();
    }
    const float lse = mx + logf(red[0]);
    for (int j = tid; j < V; j += 1024) out[j] = logits[j] - lse;
}

__global__ void zero_kernel(float* __restrict__ p, int n)
{
    const int i = blockIdx.x * blockDim.x + threadIdx.x;
    if (i < n) p[i] = 0.f;
}

// ---------------------------------------------------------------------------
extern "C" void kernel_launch(void* const* d_in, const int* in_sizes, int n_in,
                              void* d_out, int out_size, void* d_ws, size_t ws_size,
                              hipStream_t stream)
{
    const float* x          = (const float*)d_in[0];   // [B,N,D]
    const int*   edge_index = (const int*)  d_in[1];   // [B,R,2,E]
    const float* conv_W     = (const float*)d_in[2];   // [R,D,D]
    const float* W0         = (const float*)d_in[3];   // [D,D]
    const float* gate_W     = (const float*)d_in[4];   // [N*D,D]
    const float* gate_U     = (const float*)d_in[5];   // [D,D]
    const float* lin_g_W    = (const float*)d_in[6];   // [D,VG]
    const float* lin_g_b    = (const float*)d_in[7];   // [VG]
    const float* lin_s_W    = (const float*)d_in[8];   // [D,VS]
    const float* lin_s_b    = (const float*)d_in[9];   // [VS]

    float* out = (float*)d_out;
    float* pg  = out;                       // [B,VG]
    float* ps  = out + (size_t)kB * kVG;    // [B,VS]

    // Workspace layout (floats): ~4.82 MB total
    float* wsf      = (float*)d_ws;
    float* H        = wsf;                          // 6*N*D
    float* proposed = H + (size_t)kG * kN * kD;     // N*D
    float* memb     = proposed + (size_t)kN * kD;   // N*D
    float* partial  = memb + (size_t)kN * kD;       // kChunks*D
    float* gate     = partial + (size_t)kChunks * kD; // D
    float* dis      = gate + kD;                    // R*N
    float* logits   = dis + kR * kN;                // max(VG,VS)

    // mem0 = zeros
    zero_kernel<<<(kN * kD + 255) / 256, 256, 0, stream>>>(memb, kN * kD);

    for (int t = 0; t < kB; ++t) {
        const float* xs = x + (size_t)t * kN * kD;
        const int*   ei = edge_index + (size_t)t * kR * 2 * kE;

        gemm6_wmma_kernel<<<kGemmBlocks, 32 * kWavesPerBlk, 0, stream>>>(
            xs, conv_W, W0, H);
        degree_kernel<<<kR, kN, 0, stream>>>(ei, dis);
        aggregate_kernel<<<kN, kD, 0, stream>>>(ei, dis, H, proposed);

        gate_partial_kernel<<<kChunks, kChunkRows, 0, stream>>>(xs, gate_W, partial);
        gate_finalize_kernel<<<1, kD, 0, stream>>>(partial, memb, gate_U, gate);

        mem_update_kernel<<<(kN * kD) / 256, 256, 0, stream>>>(gate, proposed, memb);

        logits_kernel<<<(kVG + 255) / 256, 256, 0, stream>>>(
            memb, lin_g_W, lin_g_b, logits, kVG);
        logsoftmax_kernel<<<1, 1024, 0, stream>>>(logits, pg + (size_t)t * kVG, kVG);

        logits_kernel<<<(kVS + 255) / 256, 256, 0, stream>>>(
            memb, lin_s_W, lin_s_b, logits, kVS);
        logsoftmax_kernel<<<1, 1024, 0, stream>>>(logits, ps + (size_t)t * kVS, kVS);
    }
}